// CustomHAN_82454782148687
// MI455X (gfx1250) — compile-verified
//
#include <hip/hip_runtime.h>
#include <math.h>

typedef __attribute__((ext_vector_type(2))) float v2f;
typedef __attribute__((ext_vector_type(8))) float v8f;

// ---------------------------------------------------------------------------
// GEMM: C[M x Nout] = act(A[M x K] @ W[K x Nout] + bias)
// act: 0 = none, 1 = relu, 2 = tanh.  One wave computes one 16x16 C tile with
// V_WMMA_F32_16X16X4_F32 (fp32 A/B/C, K-step 4).  Requires M%16==0, K%4==0,
// Nout%16==0 (all true here: M=50000=3125*16, K in {128,256}, Nout in {256,128}).
// Tile-guard branch is wave-uniform -> EXEC is all-ones at every WMMA.
// ---------------------------------------------------------------------------
__global__ void han_gemm_wmma_f32(const float* __restrict__ A,
                                  const float* __restrict__ W,
                                  const float* __restrict__ bias,
                                  float* __restrict__ C,
                                  int M, int K, int Nout, int act) {
  int wid  = (blockIdx.x * blockDim.x + threadIdx.x) >> 5;  // global wave id
  int lane = threadIdx.x & 31;
  int tiles_n = Nout >> 4;
  int tiles_m = M >> 4;
  if (wid >= tiles_m * tiles_n) return;  // uniform per wave
  int tm = wid / tiles_n;
  int tn = wid - tm * tiles_n;
  int row = tm << 4, col = tn << 4;
  int li = lane & 15, half = lane >> 4;

  // A fragment 16x4 fp32: lane L holds A[row + (L&15)][k + (L>>4)*2 + j] in a[j]
  const float* Arow = A + (size_t)(row + li) * K + half * 2;
  // B fragment 4x16 fp32: lane L holds W[k + (L>>4)*2 + j][col + (L&15)] in b[j]
  const float* Wcol = W + col + li;

  v8f acc = {};
  for (int k = 0; k < K; k += 4) {
    v2f a = *(const v2f*)(Arow + k);  // 8B-aligned: row*K and half*2+k are even
    int kb = k + half * 2;
    v2f b;
    b[0] = Wcol[(size_t)(kb + 0) * Nout];
    b[1] = Wcol[(size_t)(kb + 1) * Nout];
    acc = __builtin_amdgcn_wmma_f32_16x16x4_f32(
        /*neg_a=*/false, a, /*neg_b=*/false, b,
        /*c_mod=*/(short)0, acc, /*reuse_a=*/false, /*reuse_b=*/false);
  }

  float bv = bias ? bias[col + li] : 0.0f;
#pragma unroll
  for (int r = 0; r < 8; ++r) {
    // C/D layout: VGPR r, lanes 0-15 -> M=r, lanes 16-31 -> M=r+8
    int rr = row + (half << 3) + r;
    float v = acc[r] + bv;
    if (act == 1)      v = fmaxf(v, 0.0f);
    else if (act == 2) v = tanhf(v);
    C[(size_t)rr * Nout + col + li] = v;
  }
}

// ---------------------------------------------------------------------------
// Elementwise fill
// ---------------------------------------------------------------------------
__global__ void han_fill(float* __restrict__ p, float v, int n) {
  int t = blockIdx.x * blockDim.x + threadIdx.x;
  if (t < n) p[t] = v;
}

// ---------------------------------------------------------------------------
// Per-(node, head) attention logits: s_src[n,h] = <hp[n,h,:], a_src[h,:]>,
// s_dst likewise.  H=8, DH=32, feature dim 256.
// ---------------------------------------------------------------------------
__global__ void han_head_scores(const float* __restrict__ hp,
                                const float* __restrict__ a_src,
                                const float* __restrict__ a_dst,
                                float* __restrict__ s_src,
                                float* __restrict__ s_dst, int Nn) {
  int t = blockIdx.x * blockDim.x + threadIdx.x;
  if (t >= Nn * 8) return;
  int n = t >> 3, h = t & 7;
  const float* v  = hp + (size_t)n * 256 + h * 32;
  const float* as = a_src + h * 32;
  const float* ad = a_dst + h * 32;
  float ss = 0.0f, sd = 0.0f;
#pragma unroll 8
  for (int d = 0; d < 32; ++d) {
    float x = v[d];
    ss = fmaf(x, as[d], ss);
    sd = fmaf(x, ad[d], sd);
  }
  s_src[t] = ss;
  s_dst[t] = sd;
}

__device__ __forceinline__ float han_lrelu(float x) {
  return x > 0.0f ? x : 0.2f * x;
}

// Edge index e in [0,E): real edge; e in [E, E+Nn): self loop (src=dst=e-E).
// Pass 1: segment max of leaky-relu logits (per head) via float atomicMax.
__global__ void han_edge_max(const int* __restrict__ src,
                             const int* __restrict__ dst,
                             const float* __restrict__ s_src,
                             const float* __restrict__ s_dst,
                             float* __restrict__ m, int E, int Nn) {
  int t = blockIdx.x * blockDim.x + threadIdx.x;
  int Etot = E + Nn;
  if (t >= Etot * 8) return;
  int e = t >> 3, h = t & 7;
  int s_, d_;
  if (e < E) { s_ = src[e]; d_ = dst[e]; } else { s_ = d_ = e - E; }
  float x = han_lrelu(s_src[s_ * 8 + h] + s_dst[d_ * 8 + h]);
  atomicMax(&m[d_ * 8 + h], x);
}

// Pass 2: denom[dst,h] += exp(logit - m[dst,h])
__global__ void han_edge_sum(const int* __restrict__ src,
                             const int* __restrict__ dst,
                             const float* __restrict__ s_src,
                             const float* __restrict__ s_dst,
                             const float* __restrict__ m,
                             float* __restrict__ denom, int E, int Nn) {
  int t = blockIdx.x * blockDim.x + threadIdx.x;
  int Etot = E + Nn;
  if (t >= Etot * 8) return;
  int e = t >> 3, h = t & 7;
  int s_, d_;
  if (e < E) { s_ = src[e]; d_ = dst[e]; } else { s_ = d_ = e - E; }
  float x  = han_lrelu(s_src[s_ * 8 + h] + s_dst[d_ * 8 + h]);
  float ex = __expf(x - m[d_ * 8 + h]);
  atomicAdd(&denom[d_ * 8 + h], ex);
}

// Pass 3: one wave per edge.  Lane t handles features h*32+t (h=0..7), i.e.
// head h element t -> fully coalesced 128B gathers of hp[src] rows and 128B
// atomic-add scatters into out[dst] rows.  Alphas are wave-uniform scalars.
__global__ void han_edge_aggregate(const int* __restrict__ src,
                                   const int* __restrict__ dst,
                                   const float* __restrict__ s_src,
                                   const float* __restrict__ s_dst,
                                   const float* __restrict__ m,
                                   const float* __restrict__ denom,
                                   const float* __restrict__ hp,
                                   float* __restrict__ out, int E, int Nn) {
  int wave = (blockIdx.x * blockDim.x + threadIdx.x) >> 5;
  int lane = threadIdx.x & 31;
  int Etot = E + Nn;
  if (wave >= Etot) return;  // uniform per wave
  int s_, d_;
  if (wave < E) { s_ = src[wave]; d_ = dst[wave]; } else { s_ = d_ = wave - E; }

  float alpha[8];
#pragma unroll
  for (int h = 0; h < 8; ++h) {
    float x = han_lrelu(s_src[s_ * 8 + h] + s_dst[d_ * 8 + h]);
    alpha[h] = __expf(x - m[d_ * 8 + h]) / denom[d_ * 8 + h];
  }
  const float* hs = hp + (size_t)s_ * 256;
  float* od = out + (size_t)d_ * 256;
#pragma unroll
  for (int h = 0; h < 8; ++h) {
    atomicAdd(&od[h * 32 + lane], alpha[h] * hs[h * 32 + lane]);
  }
}

// h_out = elu(agg + bias)   (bias broadcast over feature dim 256)
__global__ void han_bias_elu(const float* __restrict__ in,
                             const float* __restrict__ b,
                             float* __restrict__ out, int Nn) {
  int t = blockIdx.x * blockDim.x + threadIdx.x;
  if (t >= Nn * 256) return;
  float v = in[t] + b[t & 255];
  out[t] = v > 0.0f ? v : (__expf(v) - 1.0f);
}

// scores[n,p] = <t1[n,:], W2[:,0]>   (t1 = tanh(emb_p @ sem_W1 + b1), [N,128])
__global__ void han_sem_score(const float* __restrict__ t1,
                              const float* __restrict__ W2,
                              float* __restrict__ scores, int Nn, int p) {
  int n = blockIdx.x * blockDim.x + threadIdx.x;
  if (n >= Nn) return;
  const float* r = t1 + (size_t)n * 128;
  float s = 0.0f;
#pragma unroll 8
  for (int j = 0; j < 128; ++j) s = fmaf(r[j], W2[j], s);
  scores[n * 3 + p] = s;
}

// z[n,f] = sum_p softmax(scores[n,:])_p * embs[p][n][f]
__global__ void han_sem_fuse(const float* __restrict__ embs,
                             const float* __restrict__ scores,
                             float* __restrict__ z, int Nn) {
  int t = blockIdx.x * blockDim.x + threadIdx.x;
  if (t >= Nn * 256) return;
  int n = t >> 8, f = t & 255;
  float s0 = scores[n * 3 + 0];
  float s1 = scores[n * 3 + 1];
  float s2 = scores[n * 3 + 2];
  float mx = fmaxf(s0, fmaxf(s1, s2));
  float e0 = __expf(s0 - mx), e1 = __expf(s1 - mx), e2 = __expf(s2 - mx);
  float inv = 1.0f / (e0 + e1 + e2);
  size_t stride = (size_t)Nn * 256;
  size_t o = (size_t)n * 256 + f;
  z[t] = (e0 * embs[o] + e1 * embs[stride + o] + e2 * embs[2 * stride + o]) * inv;
}

// logits[n, 0..1] = hcls[n,:] @ cls_W2[128,2] + b2
__global__ void han_cls_out(const float* __restrict__ hcls,
                            const float* __restrict__ W2,
                            const float* __restrict__ b2,
                            float* __restrict__ out, int Nn) {
  int n = blockIdx.x * blockDim.x + threadIdx.x;
  if (n >= Nn) return;
  const float* r = hcls + (size_t)n * 128;
  float a0 = b2[0], a1 = b2[1];
#pragma unroll 8
  for (int j = 0; j < 128; ++j) {
    float x = r[j];
    a0 = fmaf(x, W2[j * 2 + 0], a0);
    a1 = fmaf(x, W2[j * 2 + 1], a1);
  }
  out[n * 2 + 0] = a0;
  out[n * 2 + 1] = a1;
}

// ---------------------------------------------------------------------------
// Host orchestration
// ---------------------------------------------------------------------------
static inline int ceil_div(long long a, int b) { return (int)((a + b - 1) / b); }

extern "C" void kernel_launch(void* const* d_in, const int* in_sizes, int n_in,
                              void* d_out, int out_size, void* d_ws, size_t ws_size,
                              hipStream_t stream) {
  const float* x       = (const float*)d_in[0];
  const int*   edges   = (const int*)  d_in[1];   // [3,2,E]
  const float* proj_W  = (const float*)d_in[2];
  const float* proj_b  = (const float*)d_in[3];
  const float* gat_W   = (const float*)d_in[4];   // [3,2,256,256]
  const float* gat_as  = (const float*)d_in[5];   // [3,2,8,32]
  const float* gat_ad  = (const float*)d_in[6];
  const float* gat_b   = (const float*)d_in[7];   // [3,2,256]
  const float* sem_W1  = (const float*)d_in[8];   // [256,128]
  const float* sem_b1  = (const float*)d_in[9];
  const float* sem_W2  = (const float*)d_in[10];  // [128,1]
  const float* cls_W1  = (const float*)d_in[11];  // [256,128]
  const float* cls_b1  = (const float*)d_in[12];
  const float* cls_W2  = (const float*)d_in[13];  // [128,2]
  const float* cls_b2  = (const float*)d_in[14];

  const int Nn = in_sizes[0] / 128;       // 50000
  const int E  = in_sizes[1] / 6;         // 800000
  const int Etot = E + Nn;

  float* ws = (float*)d_ws;
  float* h0     = ws; ws += (size_t)Nn * 256;
  float* ping   = ws; ws += (size_t)Nn * 256;
  float* pong   = ws; ws += (size_t)Nn * 256;
  float* embs   = ws; ws += (size_t)3 * Nn * 256;
  float* ssrc   = ws; ws += (size_t)Nn * 8;
  float* sdst   = ws; ws += (size_t)Nn * 8;
  float* mbuf   = ws; ws += (size_t)Nn * 8;
  float* dbuf   = ws; ws += (size_t)Nn * 8;
  float* scores = ws; ws += (size_t)Nn * 3;

  const int TB = 256;                 // 8 waves per block
  auto gemm = [&](const float* A, const float* W, const float* b, float* C,
                  int M, int K, int No, int act) {
    int tiles = (M >> 4) * (No >> 4);
    int blocks = ceil_div(tiles, TB / 32);
    han_gemm_wmma_f32<<<blocks, TB, 0, stream>>>(A, W, b, C, M, K, No, act);
  };

  // h0 = x @ proj_W + proj_b
  gemm(x, proj_W, proj_b, h0, Nn, 128, 256, 0);

  for (int p = 0; p < 3; ++p) {
    const int* srcp = edges + (size_t)p * 2 * E;
    const int* dstp = srcp + E;
    const float* hin = h0;
    for (int l = 0; l < 2; ++l) {
      int idx = p * 2 + l;
      const float* W  = gat_W  + (size_t)idx * 256 * 256;
      const float* as = gat_as + (size_t)idx * 256;
      const float* ad = gat_ad + (size_t)idx * 256;
      const float* gb = gat_b  + (size_t)idx * 256;
      float* hpb = ping;   // hin is h0 (l=0) or pong (l=1), never ping
      float* agg = pong;   // pong is dead after the GEMM reads hin

      gemm(hin, W, nullptr, hpb, Nn, 256, 256, 0);
      han_head_scores<<<ceil_div((long long)Nn * 8, TB), TB, 0, stream>>>(
          hpb, as, ad, ssrc, sdst, Nn);
      han_fill<<<ceil_div(Nn * 8, TB), TB, 0, stream>>>(mbuf, -3.0e38f, Nn * 8);
      han_fill<<<ceil_div(Nn * 8, TB), TB, 0, stream>>>(dbuf, 0.0f, Nn * 8);
      han_fill<<<ceil_div((long long)Nn * 256, TB), TB, 0, stream>>>(agg, 0.0f, Nn * 256);
      han_edge_max<<<ceil_div((long long)Etot * 8, TB), TB, 0, stream>>>(
          srcp, dstp, ssrc, sdst, mbuf, E, Nn);
      han_edge_sum<<<ceil_div((long long)Etot * 8, TB), TB, 0, stream>>>(
          srcp, dstp, ssrc, sdst, mbuf, dbuf, E, Nn);
      han_edge_aggregate<<<ceil_div((long long)Etot * 32, TB), TB, 0, stream>>>(
          srcp, dstp, ssrc, sdst, mbuf, dbuf, hpb, agg, E, Nn);

      float* hout = (l == 1) ? (embs + (size_t)p * Nn * 256) : pong;
      han_bias_elu<<<ceil_div((long long)Nn * 256, TB), TB, 0, stream>>>(agg, gb, hout, Nn);
      hin = pong;
    }
  }

  // Semantic attention
  for (int p = 0; p < 3; ++p) {
    gemm(embs + (size_t)p * Nn * 256, sem_W1, sem_b1, ping, Nn, 256, 128, 2);
    han_sem_score<<<ceil_div(Nn, TB), TB, 0, stream>>>(ping, sem_W2, scores, Nn, p);
  }
  han_sem_fuse<<<ceil_div((long long)Nn * 256, TB), TB, 0, stream>>>(embs, scores, pong, Nn);

  // Classifier
  gemm(pong, cls_W1, cls_b1, ping, Nn, 256, 128, 1);
  han_cls_out<<<ceil_div(Nn, TB), TB, 0, stream>>>(ping, cls_W2, cls_b2, (float*)d_out, Nn);
}